// MotionLoss_21534966022281
// MI455X (gfx1250) — compile-verified
//
#include <hip/hip_runtime.h>

// ---------------------------------------------------------------------------
// MotionLoss surrogate for MI455X (gfx1250, wave32).
//
// Bandwidth-bound: 52.4 MB streamed, ~34 MFLOP total -> HBM roofline ~2.3 us.
// Strategy: float2 (b64) loads of the 72-byte AoS records, scalar VALU math,
// per-wave reduction via V_WMMA_F32_16X16X4_F32 ones-matrix trick (CDNA5 path),
// deterministic two-pass block reduction through d_ws.
// ---------------------------------------------------------------------------

typedef __attribute__((ext_vector_type(2))) float v2f;
typedef __attribute__((ext_vector_type(8))) float v8f;

#define NPOINTS (512 * 1024)
#define BLOCKS  1024
#define TPB     256
#define NWAVES  (TPB / 32)

// Sum a value across all 32 lanes of the wave using one f32 WMMA.
// A (16x4): lane L puts v in VGPR0 (K=0 for lanes 0-15, K=2 for lanes 16-31),
//           0 in VGPR1 (K=1 / K=3).
// B (4x16): all ones.
// => D[m,n] = v[m] + v[m+16]  (identical for every column n).
// Lane n<16 holds D[0..7, n] in c[0..7]; lane n>=16 holds D[8..15, n-16].
// Sum the 8 accumulators, then fold halves with shfl_xor(16).
__device__ __forceinline__ float wave_sum_wmma(float v) {
  v2f a; a[0] = v;    a[1] = 0.0f;
  v2f b; b[0] = 1.0f; b[1] = 1.0f;
  v8f c = {};
  // 8 args: (neg_a, A, neg_b, B, c_mod, C, reuse_a, reuse_b)
  c = __builtin_amdgcn_wmma_f32_16x16x4_f32(false, a, false, b,
                                            (short)0, c, false, false);
  float t = ((c[0] + c[1]) + (c[2] + c[3])) + ((c[4] + c[5]) + (c[6] + c[7]));
  t += __shfl_xor(t, 16, 32);
  return t;  // full 32-lane sum, valid in every lane
}

__global__ void __launch_bounds__(TPB)
motion_loss_partial(const float* __restrict__ in,   // [N,18]
                    const float* __restrict__ tg,   // [N,7]
                    float* __restrict__ partial) {  // [BLOCKS]
  __shared__ float sdata[NWAVES];

  const int tid    = blockIdx.x * TPB + threadIdx.x;
  const int stride = gridDim.x * TPB;

  float acc = 0.0f;

  for (int i = tid; i < NPOINTS; i += stride) {
    const float2* p2 = (const float2*)in + (size_t)i * 9;  // 72B record, 8B aligned
    __builtin_prefetch(p2 + (size_t)stride * 9, 0, 3);     // global_prefetch_b8

    const float2 f01 = p2[0], f23 = p2[1], f45 = p2[2];
    const float2 f67 = p2[3], f89 = p2[4], fAB = p2[5];
    const float2 fCD = p2[6], fEF = p2[7], fGH = p2[8];

    // inputs layout: [0:3]=ploc, [3:9]=straightness s[3][2], [9:18]=ang[3][3]
    const float p0 = f01.x, p1 = f01.y, p2c = f23.x;
    const float s00 = f23.y, s01 = f45.x;
    const float s10 = f45.y, s11 = f67.x;
    const float s20 = f67.y, s21 = f89.x;
    const float ex0 = f89.y, ey0 = fAB.x, ez0 = fAB.y;   // ang[0]
    const float ex1 = fCD.x, ey1 = fCD.y, ez1 = fEF.x;   // ang[1]
    const float ex2 = fEF.y, ey2 = fGH.x, ez2 = fGH.y;   // ang[2]

    const float* t = tg + (size_t)i * 7;
    const float t0 = t[0], t1 = t[1], t2 = t[2], t3 = t[3];
    const float x = t[4], y = t[5], z = t[6];

    // dX = (p0, s00, s01); dY = (s10, p1, s11); dZ = (s20, s21, p2c)
    // M*P = MX*(MY*(MZ*P)), htm rows: [1,-ez,ey,d0],[ez,1,-ex,d1],[-ey,ex,1,d2]
    const float qx =  x   - ez2 * y + ey2 * z + s20;
    const float qy =  ez2 * x + y   - ex2 * z + s21;
    const float qz = -ey2 * x + ex2 * y + z   + p2c;

    const float rx =  qx   - ez1 * qy + ey1 * qz + s10;
    const float ry =  ez1 * qx + qy   - ex1 * qz + p1;
    const float rz = -ey1 * qx + ex1 * qy + qz   + s11;

    const float ux =  rx   - ez0 * ry + ey0 * rz + p0;
    const float uy =  ez0 * rx + ry   - ex0 * rz + s00;
    const float uz = -ey0 * rx + ex0 * ry + rz   + s01;

    const float dx = ux - x, dy = uy - y, dz = uz - z;
    const float err = sqrtf(fmaf(dx, dx, fmaf(dy, dy, fmaf(dz, dz, 1e-12f))));

    const float e0 = err - t0, e1 = dx - t1, e2 = dy - t2, e3 = dz - t3;
    acc += fmaf(e0, e0, fmaf(e1, e1, fmaf(e2, e2, e3 * e3)));
  }

  // Per-wave reduction via WMMA (EXEC is all-1s here: uniform loop, no divergence)
  const float wsum = wave_sum_wmma(acc);

  const int lane = threadIdx.x & 31;
  const int wave = threadIdx.x >> 5;
  if (lane == 0) sdata[wave] = wsum;
  __syncthreads();

  if (threadIdx.x == 0) {
    float s = 0.0f;
    for (int w = 0; w < NWAVES; ++w) s += sdata[w];   // fixed order: deterministic
    partial[blockIdx.x] = s;
  }
}

__global__ void __launch_bounds__(TPB)
motion_loss_final(const float* __restrict__ partial,  // [BLOCKS]
                  float* __restrict__ out) {          // [1]
  __shared__ float sdata[NWAVES];

  float acc = 0.0f;
  for (int i = threadIdx.x; i < BLOCKS; i += TPB) acc += partial[i];

  const float wsum = wave_sum_wmma(acc);

  const int lane = threadIdx.x & 31;
  const int wave = threadIdx.x >> 5;
  if (lane == 0) sdata[wave] = wsum;
  __syncthreads();

  if (threadIdx.x == 0) {
    float s = 0.0f;
    for (int w = 0; w < NWAVES; ++w) s += sdata[w];
    out[0] = s * (1.0f / ((float)NPOINTS * 4.0f));   // mean over [B,T,4]
  }
}

extern "C" void kernel_launch(void* const* d_in, const int* in_sizes, int n_in,
                              void* d_out, int out_size, void* d_ws, size_t ws_size,
                              hipStream_t stream) {
  const float* in = (const float*)d_in[0];   // inputs  [512,1024,18] f32
  const float* tg = (const float*)d_in[1];   // targets [512,1024,7]  f32
  float* out = (float*)d_out;                // scalar f32
  float* partial = (float*)d_ws;             // BLOCKS floats of scratch (4 KB)

  motion_loss_partial<<<BLOCKS, TPB, 0, stream>>>(in, tg, partial);
  motion_loss_final<<<1, TPB, 0, stream>>>(partial, out);
}